// CombineGraph_11501922419033
// MI455X (gfx1250) — compile-verified
//
#include <hip/hip_runtime.h>
#include <hip/hip_bf16.h>

// ---------------------------------------------------------------------------
// CombineGraph for MI455X (gfx1250, wave32).
// - fp32 GEMMs on v_wmma_f32_16x16x4_f32, register-blocked: each wave holds
//   multiple 16x16 accumulator tiles so A/B fragments are reused 2-4x and
//   live-register count stays well under 256 (no spills, no VGPR-MSB window).
// - Neighbor/h tiles staged into LDS with GLOBAL_LOAD_ASYNC_TO_LDS_B128
//   (ASYNCcnt) instead of global->VGPR->ds_store.
// - Embedding table (25.6MB) is L2-resident (192MB): gathers fused, never
//   materialized. ~26 GFLOP vs ~80MB HBM traffic -> compute-bound.
// ---------------------------------------------------------------------------

constexpr int BB  = 64;     // batch
constexpr int SS  = 64;     // session length
constexpr int DD  = 128;    // feature dim
constexpr int SMP = 12;     // neighbor samples

typedef __attribute__((ext_vector_type(2))) float v2f;
typedef __attribute__((ext_vector_type(8))) float v8f;

// D = A(16x4) * B(4x16) + C(16x16), fp32, wave32.
__device__ __forceinline__ v8f wmma_f32_4(v2f a, v2f b, v8f c) {
    return __builtin_amdgcn_wmma_f32_16x16x4_f32(
        /*neg_a=*/false, a, /*neg_b=*/false, b,
        /*c_mod=*/(short)0, c, /*reuse_a=*/false, /*reuse_b=*/false);
}

__device__ __forceinline__ float leaky(float x) { return (x >= 0.f) ? x : 0.2f * x; }

// Async 16B copy global -> LDS (tracked by ASYNCcnt). lds_dst/gsrc are
// per-lane addresses; generic->LDS offset by truncation (aperture in [63:32]).
__device__ __forceinline__ void async_b128(void* lds_dst, const void* gsrc) {
    unsigned lds = (unsigned)(uintptr_t)lds_dst;
    unsigned long long ga = (unsigned long long)(uintptr_t)gsrc;
    asm volatile("global_load_async_to_lds_b128 %0, %1, off"
                 :: "v"(lds), "v"(ga) : "memory");
}
__device__ __forceinline__ void async_wait0() {
    asm volatile("s_wait_asynccnt 0x0" ::: "memory");
}

// ---------------------------------------------------------------------------
// n_out[r,s] = adj_all[src_idx[r], s];  w_out[r,s] = num_w[src_idx[r], s]
// ---------------------------------------------------------------------------
__global__ void k_sample(const int* __restrict__ src_idx,
                         const int* __restrict__ adj_all,
                         const float* __restrict__ num_w,
                         int* __restrict__ n_out,
                         float* __restrict__ w_out,
                         int total) {
    int i = blockIdx.x * blockDim.x + threadIdx.x;
    if (i >= total) return;
    int s = i % SMP;
    int r = i / SMP;
    long o = (long)src_idx[r] * SMP + s;
    n_out[i] = adj_all[o];
    w_out[i] = num_w[o];
}

// ---------------------------------------------------------------------------
// sum_item[b,d] = sum_s emb[item[b,s],d]*mask / sum_s mask
// ---------------------------------------------------------------------------
__global__ __launch_bounds__(DD) void k_sum_item(const int* __restrict__ item,
                                                 const int* __restrict__ mask,
                                                 const float* __restrict__ emb,
                                                 float* __restrict__ outp) {
    int b = blockIdx.x, d = threadIdx.x;
    float acc = 0.f, cnt = 0.f;
    for (int s = 0; s < SS; ++s) {
        float mf = (float)mask[b * SS + s];
        acc += emb[(long)item[b * SS + s] * DD + d] * mf;
        cnt += mf;
    }
    outp[b * DD + d] = acc / cnt;
}

// ---------------------------------------------------------------------------
// LocalAggregator: one block per batch. 128 threads = 4 waves.
// ---------------------------------------------------------------------------
__global__ __launch_bounds__(128) void k_local(const int* __restrict__ inputs,
                                               const int* __restrict__ adjm,
                                               const float* __restrict__ emb,
                                               const float* __restrict__ a0,
                                               const float* __restrict__ a1,
                                               const float* __restrict__ a2,
                                               const float* __restrict__ a3,
                                               float* __restrict__ out) {
    __shared__ float hS[SS * DD];      // 32 KB
    __shared__ float alpha[SS * SS];   // 16 KB
    __shared__ float avec[4][DD];      // 2 KB

    int b = blockIdx.x;
    int tid = threadIdx.x, lane = tid & 31, wave = tid >> 5;

    // h tile -> LDS via async copies (16B per lane)
    for (int i = tid; i < SS * (DD / 4); i += 128) {
        int s = i >> 5, d4 = i & 31;                  // DD/4 == 32
        const float* src = emb + (long)inputs[b * SS + s] * DD + d4 * 4;
        async_b128(&hS[s * DD + d4 * 4], src);
    }
    for (int d = tid; d < DD; d += 128) {
        avec[0][d] = a0[d]; avec[1][d] = a1[d];
        avec[2][d] = a2[d]; avec[3][d] = a3[d];
    }
    for (int i = tid; i < SS * SS; i += 128) alpha[i] = -9e15f;
    async_wait0();
    __syncthreads();

    const int* adjb = adjm + (long)b * SS * SS;
    int m  = lane & 15;
    int kk = (lane < 16) ? 0 : 2;
    v8f zacc = {};

    // --- e_r = leaky((h * a_r) @ h^T), mask folded into alpha write --------
    // Wave owns column tile tj = wave; 4 row tiles accumulated together so
    // the B fragment and avec pair are loaded once per K-step, reused 4x.
    {
        int tj = wave;
        for (int r = 0; r < 4; ++r) {
            v8f acc[4] = {zacc, zacc, zacc, zacc};
            #pragma unroll 8
            for (int k = 0; k < DD; k += 4) {
                v2f av, bf;
                av.x = avec[r][k + kk];
                av.y = avec[r][k + kk + 1];
                bf.x = hS[(tj * 16 + m) * DD + k + kk];   // B[k][n]=h[n][k]
                bf.y = hS[(tj * 16 + m) * DD + k + kk + 1];
                #pragma unroll
                for (int ti = 0; ti < 4; ++ti) {
                    v2f af;
                    af.x = hS[(ti * 16 + m) * DD + k + kk]     * av.x;
                    af.y = hS[(ti * 16 + m) * DD + k + kk + 1] * av.y;
                    acc[ti] = wmma_f32_4(af, bf, acc[ti]);
                }
            }
            #pragma unroll
            for (int ti = 0; ti < 4; ++ti) {
                for (int e = 0; e < 8; ++e) {
                    int row = ti * 16 + e + ((lane < 16) ? 0 : 8);
                    int col = tj * 16 + (lane & 15);
                    float v = leaky(acc[ti][e]);
                    if (adjb[row * SS + col] == r + 1) alpha[row * SS + col] = v;
                }
            }
        }
    }
    __syncthreads();

    // --- softmax over j ----------------------------------------------------
    if (tid < SS) {
        float* row = alpha + tid * SS;
        float mx = row[0];
        for (int j = 1; j < SS; ++j) mx = fmaxf(mx, row[j]);
        float sm = 0.f;
        for (int j = 0; j < SS; ++j) { float e = __expf(row[j] - mx); row[j] = e; sm += e; }
        float inv = 1.f / sm;
        for (int j = 0; j < SS; ++j) row[j] *= inv;
    }
    __syncthreads();

    // --- h_local = alpha @ h -> d_out --------------------------------------
    // Wave owns 2 column tiles (tn = 2*wave, 2*wave+1) x 4 row tiles:
    // 8 accumulators, A reused 2x, B reused 4x.
    {
        int tn0 = wave * 2;
        v8f acc[4][2] = {{zacc, zacc}, {zacc, zacc}, {zacc, zacc}, {zacc, zacc}};
        #pragma unroll 4
        for (int k = 0; k < SS; k += 4) {
            v2f bf0, bf1;
            bf0.x = hS[(k + kk) * DD + tn0 * 16 + m];
            bf0.y = hS[(k + kk + 1) * DD + tn0 * 16 + m];
            bf1.x = hS[(k + kk) * DD + (tn0 + 1) * 16 + m];
            bf1.y = hS[(k + kk + 1) * DD + (tn0 + 1) * 16 + m];
            #pragma unroll
            for (int ti = 0; ti < 4; ++ti) {
                v2f af;
                af.x = alpha[(ti * 16 + m) * SS + k + kk];
                af.y = alpha[(ti * 16 + m) * SS + k + kk + 1];
                acc[ti][0] = wmma_f32_4(af, bf0, acc[ti][0]);
                acc[ti][1] = wmma_f32_4(af, bf1, acc[ti][1]);
            }
        }
        #pragma unroll
        for (int ti = 0; ti < 4; ++ti)
            #pragma unroll
            for (int j = 0; j < 2; ++j)
                for (int e = 0; e < 8; ++e) {
                    int row = ti * 16 + e + ((lane < 16) ? 0 : 8);
                    int col = (tn0 + j) * 16 + (lane & 15);
                    out[((long)b * SS + row) * DD + col] = acc[ti][j][e];
                }
    }
}

// ---------------------------------------------------------------------------
// GlobalAggregator layer. One block handles 4 "m" rows (48 neighbor rows).
// Self/neighbor features either dense or gathered from embedding on the fly.
// ---------------------------------------------------------------------------
__global__ __launch_bounds__(128, 4) void k_glayer(const float* __restrict__ ev_self,
                                                   const int* __restrict__ self_idx,
                                                   const float* __restrict__ neigh_dense,
                                                   const int* __restrict__ neigh_idx,
                                                   const float* __restrict__ emb,
                                                   const float* __restrict__ wv,
                                                   const float* __restrict__ qsum,
                                                   const float* __restrict__ w1,
                                                   const float* __restrict__ w2,
                                                   const float* __restrict__ w3,
                                                   float* __restrict__ out,
                                                   int M, int accumulate) {
    __shared__ float nb[48 * DD];     // neighbor tile, 24 KB (async-filled)
    __shared__ float amat[48 * DD];   // leaky(X@w1 + bias); reused as cat tile
    __shared__ float qv[DD];
    __shared__ float wvs[48];
    __shared__ float sc[48];

    int chunks = M / 4;
    int b  = blockIdx.x / chunks;
    int m0 = (blockIdx.x % chunks) * 4;
    int tid = threadIdx.x, lane = tid & 31, wave = tid >> 5;
    int M12 = M * SMP;

    // neighbor tile -> LDS via async global->LDS (rows hit L2: emb resident)
    for (int i = tid; i < 48 * (DD / 4); i += 128) {
        int r = i >> 5, d4 = i & 31;                  // DD/4 == 32
        long gr = (long)b * M12 + m0 * SMP + r;
        const float* src = neigh_idx ? (emb + (long)neigh_idx[gr] * DD)
                                     : (neigh_dense + gr * DD);
        async_b128(&nb[r * DD + d4 * 4], src + d4 * 4);
    }
    for (int d = tid; d < DD; d += 128) qv[d] = qsum[b * DD + d];
    if (tid < 48) wvs[tid] = wv[(long)b * M12 + m0 * SMP + tid];
    async_wait0();
    __syncthreads();

    int m  = lane & 15;
    int kk = (lane < 16) ? 0 : 2;
    v8f zacc = {};

    // --- a = leaky((q*neigh)@w1[:D] + w*w1[D]) -----------------------------
    // Wave owns 2 column tiles (tn = wave, wave+4) x 3 row tiles:
    // 6 accumulators, B reused 3x, A (q-scaled) reused 2x.
    {
        v8f acc[3][2] = {{zacc, zacc}, {zacc, zacc}, {zacc, zacc}};
        #pragma unroll 8
        for (int k = 0; k < DD; k += 4) {
            v2f qf, bf0, bf1;
            qf.x = qv[k + kk];
            qf.y = qv[k + kk + 1];
            bf0.x = w1[(k + kk) * DD + wave * 16 + m];
            bf0.y = w1[(k + kk + 1) * DD + wave * 16 + m];
            bf1.x = w1[(k + kk) * DD + (wave + 4) * 16 + m];
            bf1.y = w1[(k + kk + 1) * DD + (wave + 4) * 16 + m];
            #pragma unroll
            for (int ti = 0; ti < 3; ++ti) {
                v2f af;
                af.x = nb[(ti * 16 + m) * DD + k + kk]     * qf.x;
                af.y = nb[(ti * 16 + m) * DD + k + kk + 1] * qf.y;
                acc[ti][0] = wmma_f32_4(af, bf0, acc[ti][0]);
                acc[ti][1] = wmma_f32_4(af, bf1, acc[ti][1]);
            }
        }
        #pragma unroll
        for (int ti = 0; ti < 3; ++ti)
            #pragma unroll
            for (int j = 0; j < 2; ++j) {
                int tn = wave + j * 4;
                for (int e = 0; e < 8; ++e) {
                    int row = ti * 16 + e + ((lane < 16) ? 0 : 8);
                    int col = tn * 16 + (lane & 15);
                    amat[row * DD + col] = leaky(acc[ti][j][e] + wvs[row] * w1[DD * DD + col]);
                }
            }
    }
    __syncthreads();

    // --- scores = a @ w2, softmax over the 12 samples ----------------------
    if (tid < 48) {
        float s = 0.f;
        for (int e = 0; e < DD; ++e) s += amat[tid * DD + e] * w2[e];
        sc[tid] = s;
    }
    __syncthreads();
    if (tid < 4) {
        float mx = sc[tid * SMP];
        for (int j = 1; j < SMP; ++j) mx = fmaxf(mx, sc[tid * SMP + j]);
        float sum = 0.f;
        for (int j = 0; j < SMP; ++j) {
            float e = __expf(sc[tid * SMP + j] - mx);
            sc[tid * SMP + j] = e; sum += e;
        }
        float inv = 1.f / sum;
        for (int j = 0; j < SMP; ++j) sc[tid * SMP + j] *= inv;
    }
    __syncthreads();

    // --- cat = [self, sum_s al*neigh], zero-padded to 16 rows --------------
    float* cat = amat;                 // reuse: 16*256 = 4096 <= 48*128 floats
    for (int i = tid; i < 16 * 256; i += 128)
        if ((i >> 8) >= 4) cat[i] = 0.f;
    for (int d = tid; d < DD; d += 128) {
        for (int mm = 0; mm < 4; ++mm) {
            const float* selfp = self_idx
                ? (emb + (long)self_idx[(long)b * M + m0 + mm] * DD)
                : (ev_self + ((long)b * M + m0 + mm) * DD);
            float nv = 0.f;
            for (int s2 = 0; s2 < SMP; ++s2)
                nv += sc[mm * SMP + s2] * nb[(mm * SMP + s2) * DD + d];
            cat[mm * 256 + d]      = selfp[d];
            cat[mm * 256 + DD + d] = nv;
        }
    }
    __syncthreads();

    // --- out = relu(cat @ w3), rows 0..3 valid -----------------------------
    // Wave owns 2 column tiles (tn = wave, wave+4); A fragment reused 2x.
    {
        v8f c0 = zacc, c1 = zacc;
        #pragma unroll 8
        for (int k = 0; k < 2 * DD; k += 4) {
            v2f af, bf0, bf1;
            af.x = cat[m * 256 + k + kk];
            af.y = cat[m * 256 + k + kk + 1];
            bf0.x = w3[(k + kk) * DD + wave * 16 + m];
            bf0.y = w3[(k + kk + 1) * DD + wave * 16 + m];
            bf1.x = w3[(k + kk) * DD + (wave + 4) * 16 + m];
            bf1.y = w3[(k + kk + 1) * DD + (wave + 4) * 16 + m];
            c0 = wmma_f32_4(af, bf0, c0);
            c1 = wmma_f32_4(af, bf1, c1);
        }
        #pragma unroll
        for (int j = 0; j < 2; ++j) {
            int tn = wave + j * 4;
            for (int e = 0; e < 8; ++e) {
                int row = e + ((lane < 16) ? 0 : 8);
                if (row < 4) {
                    int col = tn * 16 + (lane & 15);
                    float v = fmaxf(j ? c1[e] : c0[e], 0.f);
                    long o = ((long)b * M + m0 + row) * DD + col;
                    if (accumulate) out[o] += v; else out[o] = v;
                }
            }
        }
    }
}

// ---------------------------------------------------------------------------
extern "C" void kernel_launch(void* const* d_in, const int* in_sizes, int n_in,
                              void* d_out, int out_size, void* d_ws, size_t ws_size,
                              hipStream_t stream) {
    const int*   inputs  = (const int*)d_in[0];
    const int*   adjm    = (const int*)d_in[1];
    const int*   mask    = (const int*)d_in[2];
    const int*   item    = (const int*)d_in[3];
    const float* emb     = (const float*)d_in[4];
    const float* a0      = (const float*)d_in[5];
    const float* a1      = (const float*)d_in[6];
    const float* a2      = (const float*)d_in[7];
    const float* a3      = (const float*)d_in[8];
    const float* g0_w1   = (const float*)d_in[9];
    const float* g0_w2   = (const float*)d_in[10];
    const float* g0_w3   = (const float*)d_in[11];
    const float* g1_w1   = (const float*)d_in[12];
    const float* g1_w2   = (const float*)d_in[13];
    const float* g1_w3   = (const float*)d_in[14];
    const int*   adj_all = (const int*)d_in[15];
    const float* num_w   = (const float*)d_in[16];
    float* outp = (float*)d_out;

    // workspace carve (~32.5 MB)
    char* w = (char*)d_ws;
    int*   n1   = (int*)w;   w += (size_t)BB * SS * SMP * 4;
    float* w1v  = (float*)w; w += (size_t)BB * SS * SMP * 4;
    int*   n2   = (int*)w;   w += (size_t)BB * SS * SMP * SMP * 4;
    float* w2v  = (float*)w; w += (size_t)BB * SS * SMP * SMP * 4;
    float* qsum = (float*)w; w += (size_t)BB * DD * 4;
    float* ev0  = (float*)w; w += (size_t)BB * SS * DD * 4;
    float* ev1  = (float*)w; // [B, S*SMP, D] = 24 MB

    int nN1 = BB * SS * SMP;
    int nN2 = BB * SS * SMP * SMP;

    k_sample<<<(nN1 + 255) / 256, 256, 0, stream>>>(inputs, adj_all, num_w, n1, w1v, nN1);
    k_sample<<<(nN2 + 255) / 256, 256, 0, stream>>>(n1, adj_all, num_w, n2, w2v, nN2);
    k_sum_item<<<BB, DD, 0, stream>>>(item, mask, emb, qsum);

    // h_local -> d_out
    k_local<<<BB, 128, 0, stream>>>(inputs, adjm, emb, a0, a1, a2, a3, outp);

    // hop 0, level 0: self=embedding[inputs], neigh=embedding[n1], M=64
    k_glayer<<<BB * 16, 128, 0, stream>>>(nullptr, inputs, nullptr, n1, emb, w1v, qsum,
                                          g0_w1, g0_w2, g0_w3, ev0, SS, 0);
    // hop 0, level 1: self=embedding[n1], neigh=embedding[n2], M=768
    k_glayer<<<BB * 192, 128, 0, stream>>>(nullptr, n1, nullptr, n2, emb, w2v, qsum,
                                           g0_w1, g0_w2, g0_w3, ev1, SS * SMP, 0);
    // hop 1, level 0: self=ev0, neigh=ev1, M=64 -> accumulate onto h_local
    k_glayer<<<BB * 16, 128, 0, stream>>>(ev0, nullptr, ev1, nullptr, emb, w1v, qsum,
                                          g1_w1, g1_w2, g1_w3, outp, SS, 1);
}